// graph_stawnet_42752104464667
// MI455X (gfx1250) — compile-verified
//
#include <hip/hip_runtime.h>
#include <math.h>

// STAWnet forward for MI455X (gfx1250, wave32).
// fp32 end-to-end; attention GEMMs on V_WMMA_F32_16X16X4_F32.

#define BB   8
#define NN   512
#define RC   32
#define DC   32
#define SC   256
#define EC   512
#define EMB  16
#define DATT 16
#define ODIM 12

typedef float v2f __attribute__((ext_vector_type(2)));
typedef float v8f __attribute__((ext_vector_type(8)));

// ---------------- start 1x1 conv: [B,2,N,13] -> [B,32,N,13] ----------------
__global__ void start_kernel(const float* __restrict__ in, const float* __restrict__ w,
                             const float* __restrict__ b, float* __restrict__ x, int total) {
  int idx = blockIdx.x * blockDim.x + threadIdx.x;
  if (idx >= total) return;
  int t = idx % 13;
  int n = (idx / 13) % NN;
  int rc = (idx / (13 * NN)) % RC;
  int bb = idx / (13 * NN * RC);
  float v = b[rc] + w[rc * 2] * in[((bb * 2 + 0) * NN + n) * 13 + t]
                  + w[rc * 2 + 1] * in[((bb * 2 + 1) * NN + n) * 13 + t];
  x[idx] = v;
}

// ------------- gated dilated temporal conv (kernel 2, VALID) ---------------
__global__ void tconv_kernel(const float* __restrict__ x, const float* __restrict__ fw,
                             const float* __restrict__ fb, const float* __restrict__ gw,
                             const float* __restrict__ gb, float* __restrict__ out,
                             int Tin, int Tc, int dil, int total) {
  int idx = blockIdx.x * blockDim.x + threadIdx.x;
  if (idx >= total) return;
  int t = idx % Tc;
  int n = (idx / Tc) % NN;
  int dc = (idx / (Tc * NN)) % DC;
  int b = idx / (Tc * NN * DC);
  float f = fb[dc], gg = gb[dc];
  const float* xb = x + ((size_t)(b * RC) * NN + n) * Tin + t;
#pragma unroll 8
  for (int rc = 0; rc < RC; ++rc) {
    float x0 = xb[(size_t)rc * NN * Tin];
    float x1 = xb[(size_t)rc * NN * Tin + dil];
    f  += fw[(dc * RC + rc) * 2] * x0 + fw[(dc * RC + rc) * 2 + 1] * x1;
    gg += gw[(dc * RC + rc) * 2] * x0 + gw[(dc * RC + rc) * 2 + 1] * x1;
  }
  out[idx] = tanhf(f) * (1.0f / (1.0f + __expf(-gg)));
}

// ---- skip: only the LAST time column of each layer survives to the head ---
__global__ void skip_kernel(const float* __restrict__ g, const float* __restrict__ sw,
                            const float* __restrict__ sb, float* __restrict__ skip,
                            int Tc, int init, int total) {
  int idx = blockIdx.x * blockDim.x + threadIdx.x;
  if (idx >= total) return;            // total = B*SC*NN, idx = (b*SC+sc)*NN+n
  int n = idx % NN;
  int sc = (idx / NN) % SC;
  int b = idx / (NN * SC);
  float acc = sb[sc];
  const float* gp = g + (size_t)(b * DC) * NN * Tc + (size_t)n * Tc + (Tc - 1);
#pragma unroll 8
  for (int c = 0; c < DC; ++c) acc += sw[sc * DC + c] * gp[(size_t)c * NN * Tc];
  if (init) skip[idx] = acc; else skip[idx] += acc;
}

// -------- q/k projection over xe = [x ; emb] (48 ch) -> [BT,N,16] ----------
__global__ void qk_kernel(const float* __restrict__ g, const float* __restrict__ sup,
                          const float* __restrict__ qw, const float* __restrict__ qb,
                          const float* __restrict__ kw, const float* __restrict__ kb,
                          float* __restrict__ Qo, float* __restrict__ Ko, int Tc, int total) {
  int idx = blockIdx.x * blockDim.x + threadIdx.x;
  if (idx >= total) return;            // idx = bt*N*16 + n*16 + d
  int d = idx % DATT;
  int n = (idx / DATT) % NN;
  int bt = idx / (DATT * NN);
  int b = bt / Tc, t = bt % Tc;
  float xe[DC + EMB];
#pragma unroll
  for (int c = 0; c < DC; ++c) xe[c] = g[(((size_t)b * DC + c) * NN + n) * Tc + t];
#pragma unroll
  for (int e = 0; e < EMB; ++e) xe[DC + e] = sup[e * NN + n];
  float qv = qb[d], kv = kb[d];
#pragma unroll
  for (int c = 0; c < DC + EMB; ++c) {
    qv += qw[d * (DC + EMB) + c] * xe[c];
    kv += kw[d * (DC + EMB) + c] * xe[c];
  }
  Qo[idx] = qv;
  Ko[idx] = kv;
}

// ------ per-(b,t) online softmax stats: row max m[v], row sum Z[v] ---------
__global__ __launch_bounds__(256) void stats_kernel(const float* __restrict__ Qg,
                                                    const float* __restrict__ Kg,
                                                    float* __restrict__ Mo,
                                                    float* __restrict__ Zo) {
  __shared__ float Ks[NN * DATT];
  const int bt = blockIdx.x;
  const int tid = threadIdx.x;
  for (int i = tid; i < NN * DATT; i += 256) Ks[i] = Kg[(size_t)bt * NN * DATT + i];
  __syncthreads();
  for (int v = tid; v < NN; v += 256) {
    float q[DATT];
#pragma unroll
    for (int d = 0; d < DATT; ++d) q[d] = Qg[(size_t)bt * NN * DATT + v * DATT + d];
    float m = -3.4e38f, Zv = 0.0f;
    for (int w = 0; w < NN; ++w) {
      float s = 0.0f;
#pragma unroll
      for (int d = 0; d < DATT; ++d) s += q[d] * Ks[w * DATT + d];
      s *= 0.25f;                       // 1/sqrt(D_ATT)
      if (s > m) { Zv = Zv * __expf(m - s) + 1.0f; m = s; }
      else       { Zv += __expf(s - m); }
    }
    Mo[(size_t)bt * NN + v] = m;
    Zo[(size_t)bt * NN + v] = Zv;
  }
}

// ---- attention apply with fp32 WMMA:  xa = (x/Z) @ exp(Q K^T/4 - m) -------
__global__ __launch_bounds__(256) void attn_kernel(const float* __restrict__ g,
                                                   const float* __restrict__ Qg,
                                                   const float* __restrict__ Kg,
                                                   const float* __restrict__ Mg,
                                                   const float* __restrict__ Zg,
                                                   float* __restrict__ xa, int Tc) {
  __shared__ float Xp[DC * NN];     // 64KB  X'[c][v] = x[c,v]/Z[v]
  __shared__ float Qs[NN * DATT];   // 32KB
  __shared__ float Ks[NN * DATT];   // 32KB
  __shared__ float Ms[NN];          // 2KB
  __shared__ float Est[8 * 256];    // 8KB   per-wave 16x16 E staging

  const int bt = blockIdx.x;
  const int b = bt / Tc, t = bt % Tc;
  const int tid = threadIdx.x;
  const float scale = 0.25f;

  for (int i = tid; i < NN * DATT; i += 256) {
    Qs[i] = Qg[(size_t)bt * NN * DATT + i];
    Ks[i] = Kg[(size_t)bt * NN * DATT + i];
  }
  for (int i = tid; i < NN; i += 256) Ms[i] = Mg[(size_t)bt * NN + i];
  for (int i = tid; i < DC * NN; i += 256) {
    int c = i / NN, v = i % NN;
    Xp[i] = g[(((size_t)b * DC + c) * NN + v) * Tc + t] / Zg[(size_t)bt * NN + v];
  }
  __syncthreads();

  const int wave = tid >> 5, lane = tid & 31;
  const int hl = lane >> 4, l16 = lane & 15;
  float* est = &Est[wave * 256];

  // Each wave owns 4 of the 32 key tiles -> uniform trip counts for barriers.
  for (int wt = wave; wt < 32; wt += 8) {
    const int wbase = wt * 16;
    v8f acc0 = {};  // channels 0..15
    v8f acc1 = {};  // channels 16..31
    for (int vb = 0; vb < 32; ++vb) {
      const int vbase = vb * 16;
      // S tile [16v x 16w] = Q_v K_w^T  (K=16 via 4x K=4 WMMAs)
      v8f s = {};
#pragma unroll
      for (int kc = 0; kc < 4; ++kc) {
        const int d0 = kc * 4 + 2 * hl;       // A/B layout: K = 2*half + vgpr
        v2f a, bf;
        a.x  = Qs[(vbase + l16) * DATT + d0];
        a.y  = Qs[(vbase + l16) * DATT + d0 + 1];
        bf.x = Ks[(wbase + l16) * DATT + d0];
        bf.y = Ks[(wbase + l16) * DATT + d0 + 1];
        s = __builtin_amdgcn_wmma_f32_16x16x4_f32(false, a, false, bf,
                                                  (short)0, s, false, false);
      }
      // E = exp(scale*S - m[v]); D layout: vgpr r -> v = vbase + r + 8*half
#pragma unroll
      for (int r = 0; r < 8; ++r) {
        const int v = vbase + r + 8 * hl;
        est[(r + 8 * hl) * 16 + l16] = __expf(s[r] * scale - Ms[v]);
      }
      __syncthreads();
      // acc += X'[c, v-chunk] @ E[v-chunk, w]  (K=16 via 4x K=4 WMMAs x 2 c-tiles)
#pragma unroll
      for (int kc = 0; kc < 4; ++kc) {
        const int k0 = kc * 4 + 2 * hl;
        v2f bf, a0, a1;
        bf.x = est[(k0) * 16 + l16];
        bf.y = est[(k0 + 1) * 16 + l16];
        a0.x = Xp[l16 * NN + vbase + k0];
        a0.y = Xp[l16 * NN + vbase + k0 + 1];
        a1.x = Xp[(16 + l16) * NN + vbase + k0];
        a1.y = Xp[(16 + l16) * NN + vbase + k0 + 1];
        acc0 = __builtin_amdgcn_wmma_f32_16x16x4_f32(false, a0, false, bf,
                                                     (short)0, acc0, false, false);
        acc1 = __builtin_amdgcn_wmma_f32_16x16x4_f32(false, a1, false, bf,
                                                     (short)0, acc1, false, false);
      }
      __syncthreads();
    }
    // Write D: lane -> (c = r + 8*half, w = wbase + l16)
#pragma unroll
    for (int r = 0; r < 8; ++r) {
      const int c0 = r + 8 * hl;
      const int w = wbase + l16;
      xa[(((size_t)b * DC + c0) * NN + w) * Tc + t] = acc0[r];
      xa[(((size_t)b * DC + 16 + c0) * NN + w) * Tc + t] = acc1[r];
    }
  }
}

// ------------- gmlp over concat(x, xa) + residual add ----------------------
__global__ void gmlp_kernel(const float* __restrict__ g, const float* __restrict__ xav,
                            const float* __restrict__ w, const float* __restrict__ bias,
                            const float* __restrict__ xres, float* __restrict__ xo,
                            int Tp, int Tc, int total) {
  int idx = blockIdx.x * blockDim.x + threadIdx.x;
  if (idx >= total) return;
  int t = idx % Tc;
  int n = (idx / Tc) % NN;
  int o = (idx / (Tc * NN)) % RC;
  int b = idx / (Tc * NN * RC);
  float acc = bias[o];
  const float* gp = g + (size_t)(b * DC) * NN * Tc + (size_t)n * Tc + t;
  const float* xp = xav + (size_t)(b * DC) * NN * Tc + (size_t)n * Tc + t;
#pragma unroll 8
  for (int c = 0; c < DC; ++c) {
    acc += w[o * 64 + c] * gp[(size_t)c * NN * Tc];
    acc += w[o * 64 + 32 + c] * xp[(size_t)c * NN * Tc];
  }
  acc += xres[(((size_t)b * RC + o) * NN + n) * Tp + (t + Tp - Tc)];
  xo[idx] = acc;
}

// ----------------- BatchNorm (training-mode batch stats) -------------------
__global__ __launch_bounds__(256) void bnstats_kernel(const float* __restrict__ x,
                                                      float* __restrict__ st, int Tc) {
  __shared__ float s1[256], s2[256];
  const int c = blockIdx.x;
  const int tid = threadIdx.x;
  const int per_b = NN * Tc;
  const int total = BB * per_b;
  float a = 0.0f, q = 0.0f;
  for (int j = tid; j < total; j += 256) {
    int b = j / per_b, r = j % per_b;
    float v = x[((size_t)(b * RC + c)) * per_b + r];
    a += v; q += v * v;
  }
  s1[tid] = a; s2[tid] = q;
  __syncthreads();
  for (int off = 128; off > 0; off >>= 1) {
    if (tid < off) { s1[tid] += s1[tid + off]; s2[tid] += s2[tid + off]; }
    __syncthreads();
  }
  if (tid == 0) {
    float mean = s1[0] / (float)total;
    float var = s2[0] / (float)total - mean * mean;   // biased variance
    st[c] = mean;
    st[RC + c] = var;
  }
}

__global__ void bnapply_kernel(float* __restrict__ x, const float* __restrict__ st,
                               const float* __restrict__ gm, const float* __restrict__ bt_,
                               int Tc, int total) {
  int idx = blockIdx.x * blockDim.x + threadIdx.x;
  if (idx >= total) return;
  int c = (idx / (NN * Tc)) % RC;
  float mean = st[c], var = st[RC + c];
  x[idx] = gm[c] * (x[idx] - mean) * rsqrtf(var + 1e-5f) + bt_[c];
}

// ------------------------- output head -------------------------------------
__global__ __launch_bounds__(256) void head_kernel(const float* __restrict__ skip,
                                                   const float* __restrict__ w1,
                                                   const float* __restrict__ b1,
                                                   const float* __restrict__ w2,
                                                   const float* __restrict__ b2,
                                                   float* __restrict__ out) {
  __shared__ float s[SC];
  __shared__ float h[EC];
  const int bn = blockIdx.x;
  const int b = bn / NN, n = bn % NN;
  const int tid = threadIdx.x;
  s[tid] = fmaxf(skip[((size_t)(b * SC + tid)) * NN + n], 0.0f);
  __syncthreads();
  for (int j = tid; j < EC; j += 256) {
    float acc = b1[j];
#pragma unroll 8
    for (int c = 0; c < SC; ++c) acc += w1[j * SC + c] * s[c];
    h[j] = fmaxf(acc, 0.0f);
  }
  __syncthreads();
  if (tid < ODIM) {
    float acc = b2[tid];
#pragma unroll 8
    for (int j = 0; j < EC; ++j) acc += w2[tid * EC + j] * h[j];
    out[((size_t)(b * ODIM + tid)) * NN + n] = acc;
  }
}

// ===========================================================================
extern "C" void kernel_launch(void* const* d_in, const int* in_sizes, int n_in,
                              void* d_out, int out_size, void* d_ws, size_t ws_size,
                              hipStream_t stream) {
  const float* input   = (const float*)d_in[0];
  const float* sup     = (const float*)d_in[1];
  const float* start_w = (const float*)d_in[2];
  const float* start_b = (const float*)d_in[3];
  const float* filt_w  = (const float*)d_in[4];
  const float* filt_b  = (const float*)d_in[5];
  const float* gate_w  = (const float*)d_in[6];
  const float* gate_b  = (const float*)d_in[7];
  const float* skip_w  = (const float*)d_in[8];
  const float* skip_b  = (const float*)d_in[9];
  const float* q_w     = (const float*)d_in[10];
  const float* q_b     = (const float*)d_in[11];
  const float* k_w     = (const float*)d_in[12];
  const float* k_b     = (const float*)d_in[13];
  const float* gmlp_w  = (const float*)d_in[14];
  const float* gmlp_b  = (const float*)d_in[15];
  const float* bn_g    = (const float*)d_in[16];
  const float* bn_b    = (const float*)d_in[17];
  const float* e1w     = (const float*)d_in[18];
  const float* e1b     = (const float*)d_in[19];
  const float* e2w     = (const float*)d_in[20];
  const float* e2b     = (const float*)d_in[21];
  float* out = (float*)d_out;
  float* ws = (float*)d_ws;

  // Workspace layout (floats), total ~9.3M floats (~37MB)
  const size_t XSZ  = (size_t)BB * RC * NN * 13;      // 1,703,936
  const size_t GSZ  = (size_t)BB * DC * NN * 12;      // 1,572,864 (max post-conv)
  const size_t SKSZ = (size_t)BB * SC * NN;           // 1,048,576
  const size_t QSZ  = (size_t)BB * 12 * NN * DATT;    //   786,432
  const size_t MSZ  = (size_t)BB * 12 * NN;           //    49,152

  float* x0  = ws;
  float* x1  = x0 + XSZ;
  float* gb  = x1 + XSZ;
  float* xab = gb + GSZ;
  float* skp = xab + GSZ;
  float* Qb  = skp + SKSZ;
  float* Kb  = Qb + QSZ;
  float* Mb  = Kb + QSZ;
  float* Zb  = Mb + MSZ;
  float* st  = Zb + MSZ;   // 64 floats (mean,var per channel)

  float* xc = x0;
  float* xn = x1;

  {
    int total = (int)XSZ;
    start_kernel<<<(total + 255) / 256, 256, 0, stream>>>(input, start_w, start_b, xc, total);
  }

  int T = 13;
  for (int i = 0; i < 8; ++i) {
    const int d = (i % 2 == 0) ? 1 : 2;
    const int Tc = T - d;
    {
      int total = BB * DC * NN * Tc;
      tconv_kernel<<<(total + 255) / 256, 256, 0, stream>>>(
          xc, filt_w + (size_t)i * DC * RC * 2, filt_b + i * DC,
          gate_w + (size_t)i * DC * RC * 2, gate_b + i * DC, gb, T, Tc, d, total);
    }
    {
      int total = BB * SC * NN;
      skip_kernel<<<(total + 255) / 256, 256, 0, stream>>>(
          gb, skip_w + (size_t)i * SC * DC, skip_b + i * SC, skp, Tc, (i == 0) ? 1 : 0, total);
    }
    {
      int total = BB * Tc * NN * DATT;
      qk_kernel<<<(total + 255) / 256, 256, 0, stream>>>(
          gb, sup, q_w + (size_t)i * DATT * (DC + EMB), q_b + i * DATT,
          k_w + (size_t)i * DATT * (DC + EMB), k_b + i * DATT, Qb, Kb, Tc, total);
    }
    stats_kernel<<<BB * Tc, 256, 0, stream>>>(Qb, Kb, Mb, Zb);
    attn_kernel<<<BB * Tc, 256, 0, stream>>>(gb, Qb, Kb, Mb, Zb, xab, Tc);
    {
      int total = BB * RC * NN * Tc;
      gmlp_kernel<<<(total + 255) / 256, 256, 0, stream>>>(
          gb, xab, gmlp_w + (size_t)i * RC * 64, gmlp_b + i * RC, xc, xn, T, Tc, total);
    }
    bnstats_kernel<<<RC, 256, 0, stream>>>(xn, st, Tc);
    {
      int total = BB * RC * NN * Tc;
      bnapply_kernel<<<(total + 255) / 256, 256, 0, stream>>>(
          xn, st, bn_g + i * RC, bn_b + i * RC, Tc, total);
    }
    float* tmp = xc; xc = xn; xn = tmp;
    T = Tc;
  }

  head_kernel<<<BB * NN, 256, 0, stream>>>(skp, e1w, e1b, e2w, e2b, out);

  (void)in_sizes; (void)n_in; (void)out_size; (void)ws_size;
}